// RecursiveMamba2_v28_43250320670990
// MI455X (gfx1250) — compile-verified
//
#include <hip/hip_runtime.h>
#include <hip/hip_bf16.h>
#include <math.h>
#include <stdint.h>

// Problem constants (from reference): logits (4,4,256,50257) f32
#define N_LOOPS  4
#define BB       4
#define LL       256
#define VV       50257
#define CHUNK    4096                       // floats per TDM chunk (16 KB)
#define NCHUNK   ((VV + CHUNK - 1) / CHUNK) // 13
#define BMW      ((VV + 31) / 32)           // 1571 bitmap words
#define NTHREADS 256
#define NWAVES   (NTHREADS / 32)            // wave32 -> 8 waves/block

#if defined(__has_builtin)
#  if __has_builtin(__builtin_amdgcn_tensor_load_to_lds) && __has_builtin(__builtin_amdgcn_s_wait_tensorcnt)
#    define HAVE_TDM 1
#  endif
#endif
#ifndef HAVE_TDM
#  define HAVE_TDM 0
#endif

typedef unsigned int u32;
typedef __attribute__((ext_vector_type(4))) u32 u32x4;
typedef __attribute__((ext_vector_type(8))) int i32x8;
typedef __attribute__((ext_vector_type(4))) int i32x4;

#if HAVE_TDM
// Build a CDNA5 TDM descriptor (D# per CDNA5 ISA) for a 1-D
// row chunk: tile_dim0 = tile_elems, tensor_dim0 = remaining elems (so the
// final chunk's tail OOB-reads return zero — we never read those anyway).
__device__ __forceinline__ void tdm_load_chunk(const float* gsrc, u32 lds_byte_addr,
                                               int remaining_elems, int tile_elems) {
  unsigned long long ga = (unsigned long long)(uintptr_t)gsrc;
  u32 td0 = (u32)remaining_elems;
  u32x4 g0;
  g0.x = 1u;                                              // count=1, user desc
  g0.y = lds_byte_addr;                                   // LDS byte address
  g0.z = (u32)(ga & 0xFFFFFFFFull);                       // global_addr[31:0]
  g0.w = (u32)((ga >> 32) & 0x1FFFFFFull) | (2u << 30);   // addr[56:32] | type=2
  i32x8 g1;
  g1[0] = (int)0x00020000u;                               // data_size=2 (4B), mask=0
  g1[1] = (int)((td0 & 0xFFFFu) << 16);                   // tensor_dim0[15:0]
  g1[2] = (int)(((td0 >> 16) & 0xFFFFu) | (1u << 16));    // tensor_dim0[31:16] | tensor_dim1=1
  g1[3] = (int)(((u32)tile_elems & 0xFFFFu) << 16);       // tile_dim0
  g1[4] = 1;                                              // tile_dim1=1, tile_dim2=0
  g1[5] = (int)td0;                                       // tensor_dim0_stride lo (unused, tile_dim1==1)
  g1[6] = 0;
  g1[7] = 0;
  i32x4 gz4 = {0, 0, 0, 0};                               // groups 2/3: dims/tiles unused
  i32x8 gz8 = {0, 0, 0, 0, 0, 0, 0, 0};                   // toolchain 6-arg extra group
  __builtin_amdgcn_tensor_load_to_lds(g0, g1, gz4, gz4, gz8, 0);
}
#endif

// One block per (loop l, batch b) row. Computes ce, acc, valid for that row.
__global__ __launch_bounds__(NTHREADS) void ce_row_kernel(
    const float* __restrict__ logits, const int* __restrict__ input_ids,
    const int* __restrict__ ans_starts, const int* __restrict__ chain_targets,
    float* __restrict__ ws)
{
  __shared__ u32   bitmap[BMW];
  __shared__ float buf[2][CHUNK];
  __shared__ float red_m[NWAVES];
  __shared__ float red_s[NWAVES];
  __shared__ int   red_i[NWAVES];
  __shared__ float sh_xtgt;

  const int tid = threadIdx.x;
  const int row = blockIdx.x;       // row = l*BB + b
  const int l   = row / BB;
  const int b   = row - l * BB;

  const int ans = ans_starts[b];
  int as_idx = ans - 1;
  as_idx = as_idx < 0 ? 0 : (as_idx > (LL - 1) ? (LL - 1) : as_idx);
  const float* rowp = logits + (((long long)(l * BB + b) * LL + as_idx) * (long long)VV);
  const int tgt = chain_targets[b * N_LOOPS + l];

#if HAVE_TDM
  // Wave 0 kicks off the first two DMA chunks immediately (TDM ignores EXEC;
  // wave-level branch ensures only one wave issues).
  if (tid < 32) {
    tdm_load_chunk(rowp, (u32)(uintptr_t)&buf[0][0], VV, CHUNK);
    tdm_load_chunk(rowp + CHUNK, (u32)(uintptr_t)&buf[1][0], VV - CHUNK, CHUNK);
  }
#endif

  // Build valid_tok bitmap for batch row b while DMA is in flight.
  for (int i = tid; i < BMW; i += NTHREADS) bitmap[i] = 0u;
  if (tid == 0) sh_xtgt = rowp[tgt];
  __syncthreads();
  {
    const int tok = input_ids[b * LL + tid];   // NTHREADS == LL
    atomicOr(&bitmap[((u32)tok) >> 5], 1u << (tok & 31));
  }
#if HAVE_TDM
  if (tid < 32) __builtin_amdgcn_s_wait_tensorcnt(1);   // chunk 0 resident
#endif
  __syncthreads();   // bitmap complete + chunk 0 visible to all waves

  // Streaming logsumexp + first-index argmax over valid vocab entries.
  float m  = -INFINITY;
  float s  = 0.0f;
  int   ai = 0x7FFFFFFF;

  for (int c = 0; c < NCHUNK; ++c) {
    const int p    = c & 1;
    const int base = c * CHUNK;
    const int lim  = (VV - base) < CHUNK ? (VV - base) : CHUNK;
#if !HAVE_TDM
    for (int i = tid; i < lim; i += NTHREADS) buf[p][i] = rowp[base + i];
    __syncthreads();
#endif
    for (int i = tid; i < lim; i += NTHREADS) {
      const int v = base + i;
      const u32 wbits = bitmap[v >> 5];
      if ((wbits >> (v & 31)) & 1u) {
        const float x = buf[p][i];
        if (x > m) {                    // strictly greater: first max index wins
          s  = s * expf(m - x) + 1.0f;  // s==0 & expf(-inf)==0 on first hit
          m  = x;
          ai = v;
        } else {
          s += expf(x - m);
        }
      }
    }
    __syncthreads();   // all waves done reading buf[p] before it is reused
#if HAVE_TDM
    if (tid < 32) {
      if (c + 2 < NCHUNK) {
        const int nb = (c + 2) * CHUNK;
        tdm_load_chunk(rowp + nb, (u32)(uintptr_t)&buf[p][0], VV - nb, CHUNK);
        __builtin_amdgcn_s_wait_tensorcnt(1);  // {c+1,c+2} in flight -> c+1 done
      } else {
        __builtin_amdgcn_s_wait_tensorcnt(0);  // drain
      }
    }
    __syncthreads();   // next buffer resident before anyone reads it
#endif
  }

  // Wave32 reduction of (m, s, argmax-idx) with reference tie-breaking.
  for (int off = 16; off > 0; off >>= 1) {
    const float m2 = __shfl_xor(m, off, 32);
    const float s2 = __shfl_xor(s, off, 32);
    const int   i2 = __shfl_xor(ai, off, 32);
    const float M  = fmaxf(m, m2);
    float sN = 0.0f;
    if (M != -INFINITY) sN = s * expf(m - M) + s2 * expf(m2 - M);
    int iN;
    if      (m2 > m)  iN = i2;
    else if (m2 == m) iN = (i2 < ai ? i2 : ai);
    else              iN = ai;
    m = M; s = sN; ai = iN;
  }
  if ((tid & 31) == 0) {
    red_m[tid >> 5] = m; red_s[tid >> 5] = s; red_i[tid >> 5] = ai;
  }
  __syncthreads();

  if (tid == 0) {
    float M = red_m[0]; float S = red_s[0]; int I = red_i[0];
    for (int w = 1; w < NWAVES; ++w) {
      const float m2 = red_m[w]; const float s2 = red_s[w]; const int i2 = red_i[w];
      const float Mn = fmaxf(M, m2);
      float Sn = 0.0f;
      if (Mn != -INFINITY) Sn = S * expf(M - Mn) + s2 * expf(m2 - Mn);
      if      (m2 > M)  I = i2;
      else if (m2 == M) I = (i2 < I ? i2 : I);
      M = Mn; S = Sn;
    }
    const bool pos_ok = (ans >= 1) && (ans < LL);
    const bool tgt_ok = ((bitmap[((u32)tgt) >> 5] >> (tgt & 31)) & 1u) != 0u;
    const bool valid  = pos_ok && tgt_ok;
    // logp[safe_tgt] = x_safe - (M + log S); invalid rows use safe_tgt = pred,
    // whose logit equals M, giving ce = log S (matches reference exactly).
    const float xsafe = valid ? sh_xtgt : M;
    const float ce    = (M + logf(S)) - xsafe;
    const float acc   = (I == tgt) ? 1.0f : 0.0f;
    ws[row * 4 + 0] = ce;
    ws[row * 4 + 1] = acc;
    ws[row * 4 + 2] = valid ? 1.0f : 0.0f;
  }
}

// Fold the 16 per-row triples into (avg_loss, avg_acc, final_loop_acc).
__global__ void ce_finalize_kernel(const float* __restrict__ ws, float* __restrict__ out)
{
  if (threadIdx.x == 0 && blockIdx.x == 0) {
    float sum_loss = 0.0f, sum_acc = 0.0f, nvalid = 0.0f, final_acc = 0.0f;
    for (int l = 0; l < N_LOOPS; ++l) {
      float cnt = 0.0f, sl = 0.0f, sa = 0.0f;
      for (int b = 0; b < BB; ++b) {
        const int r = l * BB + b;
        const float vf = ws[r * 4 + 2];
        cnt += vf;
        sl  += ws[r * 4 + 0] * vf;
        sa  += ws[r * 4 + 1] * vf;
      }
      const float denom = fmaxf(cnt, 1.0f);
      const float ll  = sl / denom;
      const float la  = sa / denom;
      const float has = (cnt > 0.0f) ? 1.0f : 0.0f;
      sum_loss += ll * has;
      sum_acc  += la * has;
      nvalid   += has;
      if (l == N_LOOPS - 1) final_acc = la;
    }
    const float nv = fmaxf(nvalid, 1.0f);
    out[0] = sum_loss / nv;
    out[1] = sum_acc / nv;
    out[2] = final_acc;
  }
}

extern "C" void kernel_launch(void* const* d_in, const int* in_sizes, int n_in,
                              void* d_out, int out_size, void* d_ws, size_t ws_size,
                              hipStream_t stream) {
  (void)in_sizes; (void)n_in; (void)out_size; (void)ws_size;
  const float* logits        = (const float*)d_in[0];
  const int*   input_ids     = (const int*)d_in[1];
  const int*   ans_starts    = (const int*)d_in[2];
  const int*   chain_targets = (const int*)d_in[3];
  float* out = (float*)d_out;
  float* ws  = (float*)d_ws;

  ce_row_kernel<<<dim3(N_LOOPS * BB), dim3(NTHREADS), 0, stream>>>(
      logits, input_ids, ans_starts, chain_targets, ws);
  ce_finalize_kernel<<<dim3(1), dim3(32), 0, stream>>>(ws, out);
}